// PINNLoss_28819230556340
// MI455X (gfx1250) — compile-verified
//
#include <hip/hip_runtime.h>
#include <math.h>

typedef float v2f __attribute__((ext_vector_type(2)));
typedef float v8f __attribute__((ext_vector_type(8)));

#define HID 128
#define PTS 16       // points per workgroup
#define NTH 128      // 4 waves of 32
#define PITCH 132    // 16-row tile pitch: lane-stride banks 4m -> conflict-free b64
#define W2P 134      // W2 pitch: lane-stride banks 6m -> conflict-free b64, both halves

// LDS layout (floats)
constexpr int SZ_W2   = HID * W2P;            // 17152
constexpr int SZ_T    = PTS * PITCH;          // 2112
constexpr int OFF_W2  = 0;                    // W2 row-major   (GEMM2 B)
constexpr int OFF_W2T = OFF_W2  + SZ_W2;      // W2 transposed  (GEMM1 B)
constexpr int OFF_H1  = OFF_W2T + SZ_W2;      // h1 (also GEMM1 A for t=0)
constexpr int OFF_A   = OFF_H1  + SZ_T;       // 4 staging tiles (GEMM1 t=1..4; reused as GEMM2 A)
constexpr int OFF_O   = OFF_A   + 4 * SZ_T;   // 5 tiles: h, J2[0..2], r
constexpr int OFF_V   = OFF_O   + 5 * SZ_T;   // 3 tiles: V[:,i]
constexpr int OFF_W1  = OFF_V   + 3 * SZ_T;   // 3x128
constexpr int OFF_WD  = OFF_W1  + 3 * HID;    // 128x3
constexpr int OFF_WP  = OFF_WD  + 3 * HID;
constexpr int OFF_WS  = OFF_WP  + HID;
constexpr int OFF_B1  = OFF_WS  + HID;
constexpr int OFF_B2  = OFF_B1  + HID;
constexpr int OFF_X   = OFF_B2  + HID;        // 16x3 coords
constexpr int OFF_RED = OFF_X   + PTS * 3;    // 128 x 8 partials
constexpr int OFF_LOSS= OFF_RED + NTH * 8;    // 16 x 4 losses
constexpr int SMEM_FLOATS = OFF_LOSS + PTS * 4;   // ~251 KB

__global__ __launch_bounds__(NTH) void pinn_main(
    const float* __restrict__ coords, const float* __restrict__ labels,
    const float* __restrict__ tstiff,
    const float* __restrict__ W1, const float* __restrict__ b1,
    const float* __restrict__ W2, const float* __restrict__ b2,
    const float* __restrict__ Wp, const float* __restrict__ bp,
    const float* __restrict__ Ws, const float* __restrict__ bs,
    const float* __restrict__ Wd,
    float* __restrict__ partial, int Bn)
{
    extern __shared__ float sm[];
    const int tid  = threadIdx.x;
    const int lane = tid & 31;
    const int wave = tid >> 5;
    const int p0   = blockIdx.x * PTS;

    // ---- stage weights: W2 both row-major and transposed ----
    for (int idx = tid; idx < HID * HID; idx += NTH) {
        int m = idx >> 7, n = idx & 127;
        float w = W2[idx];
        sm[OFF_W2  + m * W2P + n] = w;
        sm[OFF_W2T + n * W2P + m] = w;
    }
    for (int idx = tid; idx < 3 * HID; idx += NTH) {
        sm[OFF_W1 + idx] = W1[idx];
        sm[OFF_WD + idx] = Wd[idx];
    }
    for (int idx = tid; idx < HID; idx += NTH) {
        sm[OFF_WP + idx] = Wp[idx];
        sm[OFF_WS + idx] = Ws[idx];
        sm[OFF_B1 + idx] = b1[idx];
        sm[OFF_B2 + idx] = b2[idx];
    }
    if (tid < PTS * 3) sm[OFF_X + tid] = coords[p0 * 3 + tid];
    __syncthreads();

    // ---- layer 1 + GEMM1 A-tile staging ----
    for (int e = tid; e < PTS * HID; e += NTH) {
        int p = e >> 7, m = e & 127;
        float w10 = sm[OFF_W1 + m], w11 = sm[OFF_W1 + HID + m], w12 = sm[OFF_W1 + 2*HID + m];
        float z = sm[OFF_B1 + m]
                + sm[OFF_X + p * 3 + 0] * w10
                + sm[OFF_X + p * 3 + 1] * w11
                + sm[OFF_X + p * 3 + 2] * w12;
        float h1   = tanhf(z);
        float g1   = 1.f - h1 * h1;
        float g1dd = -2.f * h1 * g1;
        int o = p * PITCH + m;
        sm[OFF_H1 + o]            = h1;                                   // t=0: z2
        sm[OFF_A  + 0 * SZ_T + o] = w10 * g1;                             // t=1: J2[0]
        sm[OFF_A  + 1 * SZ_T + o] = w11 * g1;                             // t=2: J2[1]
        sm[OFF_A  + 2 * SZ_T + o] = w12 * g1;                             // t=3: J2[2]
        sm[OFF_A  + 3 * SZ_T + o] = (w10*w10 + w11*w11 + w12*w12) * g1dd; // t=4: r
    }
    __syncthreads();

    const int mrow  = lane & 15;          // A-row / B-row / C-col selector
    const int khalf = (lane >> 4) * 2;    // k sub-pair per lane half
    const int prow  = (lane >> 4) << 3;   // C-fragment row base

    // ---- GEMM1: O_t = A_t @ W2; both N-tiles interleaved (shared A load, 2 chains) ----
    for (int t = 0; t < 5; ++t) {
        const int aoff = (t == 0) ? OFF_H1 : (OFF_A + (t - 1) * SZ_T);
        const float* Ab  = &sm[aoff + mrow * PITCH];
        const int n0 = wave * 32;
        const float* Bb0 = &sm[OFF_W2T + (n0 + mrow) * W2P];
        const float* Bb1 = &sm[OFF_W2T + (n0 + 16 + mrow) * W2P];
        v8f c0 = {0.f,0.f,0.f,0.f,0.f,0.f,0.f,0.f};
        v8f c1 = {0.f,0.f,0.f,0.f,0.f,0.f,0.f,0.f};
        #pragma unroll 8
        for (int kk = 0; kk < 32; ++kk) {
            const int kc = kk * 4 + khalf;
            v2f a  = *reinterpret_cast<const v2f*>(Ab  + kc);
            v2f b0 = *reinterpret_cast<const v2f*>(Bb0 + kc);
            v2f b1 = *reinterpret_cast<const v2f*>(Bb1 + kc);
            c0 = __builtin_amdgcn_wmma_f32_16x16x4_f32(false, a, false, b0,
                                                       (short)0, c0, false, false);
            c1 = __builtin_amdgcn_wmma_f32_16x16x4_f32(false, a, false, b1,
                                                       (short)0, c1, false, false);
        }
        #pragma unroll
        for (int v = 0; v < 8; ++v) {
            int p = v + prow;
            float v0 = c0[v], v1 = c1[v];
            if (t == 0) {
                v0 = tanhf(v0 + sm[OFF_B2 + n0 + mrow]);        // h = tanh(z2+b2)
                v1 = tanhf(v1 + sm[OFF_B2 + n0 + 16 + mrow]);
            }
            sm[OFF_O + t * SZ_T + p * PITCH + n0 + mrow]      = v0;
            sm[OFF_O + t * SZ_T + p * PITCH + n0 + 16 + mrow] = v1;
        }
    }
    __syncthreads();

    // ---- stage GEMM2 A-tiles: A2[i][p,n] = g2[p,n] * Wd[n,i] (reuse OFF_A) ----
    for (int e = tid; e < PTS * HID; e += NTH) {
        int p = e >> 7, n = e & 127;
        float h  = sm[OFF_O + p * PITCH + n];
        float g2 = 1.f - h * h;
        sm[OFF_A + 0 * SZ_T + p * PITCH + n] = g2 * sm[OFF_WD + n * 3 + 0];
        sm[OFF_A + 1 * SZ_T + p * PITCH + n] = g2 * sm[OFF_WD + n * 3 + 1];
        sm[OFF_A + 2 * SZ_T + p * PITCH + n] = g2 * sm[OFF_WD + n * 3 + 2];
    }
    __syncthreads();

    // ---- GEMM2: V[p,i,m] = sum_n A2[i][p,n] * W2[m,n]; both M-tiles interleaved ----
    for (int i = 0; i < 3; ++i) {
        const float* Ab  = &sm[OFF_A + i * SZ_T + mrow * PITCH];
        const int m0 = wave * 32;
        const float* Bb0 = &sm[OFF_W2 + (m0 + mrow) * W2P];
        const float* Bb1 = &sm[OFF_W2 + (m0 + 16 + mrow) * W2P];
        v8f c0 = {0.f,0.f,0.f,0.f,0.f,0.f,0.f,0.f};
        v8f c1 = {0.f,0.f,0.f,0.f,0.f,0.f,0.f,0.f};
        #pragma unroll 8
        for (int kk = 0; kk < 32; ++kk) {
            const int kc = kk * 4 + khalf;
            v2f a  = *reinterpret_cast<const v2f*>(Ab  + kc);
            v2f b0 = *reinterpret_cast<const v2f*>(Bb0 + kc);
            v2f b1 = *reinterpret_cast<const v2f*>(Bb1 + kc);
            c0 = __builtin_amdgcn_wmma_f32_16x16x4_f32(false, a, false, b0,
                                                       (short)0, c0, false, false);
            c1 = __builtin_amdgcn_wmma_f32_16x16x4_f32(false, a, false, b1,
                                                       (short)0, c1, false, false);
        }
        #pragma unroll
        for (int v = 0; v < 8; ++v) {
            int p = v + prow;
            sm[OFF_V + i * SZ_T + p * PITCH + m0 + mrow]      = c0[v];
            sm[OFF_V + i * SZ_T + p * PITCH + m0 + 16 + mrow] = c1[v];
        }
    }
    __syncthreads();

    // ---- per-point contractions: 8 threads per point ----
    {
        const int p = tid >> 3, s = tid & 7;
        const int base = p * PITCH;
        float lap0=0,lap1=0,lap2=0, gd0=0,gd1=0,gd2=0, zp=0, zs=0;
        for (int n = s * 16; n < s * 16 + 16; ++n) {
            float h    = sm[OFF_O + base + n];
            float g2   = 1.f - h * h;
            float g2dd = -2.f * h * g2;
            float J0 = sm[OFF_O + 1 * SZ_T + base + n];
            float J1 = sm[OFF_O + 2 * SZ_T + base + n];
            float J2 = sm[OFF_O + 3 * SZ_T + base + n];
            float r  = sm[OFF_O + 4 * SZ_T + base + n];
            float q  = J0*J0 + J1*J1 + J2*J2;
            float wd0 = sm[OFF_WD + n*3], wd1 = sm[OFF_WD + n*3+1], wd2 = sm[OFF_WD + n*3+2];
            float Pn = wd0*J0 + wd1*J1 + wd2*J2;
            float tl = g2dd * q + g2 * r;
            lap0 += wd0 * tl; lap1 += wd1 * tl; lap2 += wd2 * tl;
            float gp = g2dd * Pn;
            gd0 += gp * J0; gd1 += gp * J1; gd2 += gp * J2;
            zp += h * sm[OFF_WP + n];
            zs += h * sm[OFF_WS + n];
            float h1   = sm[OFF_H1 + base + n];
            float g1dd = -2.f * h1 * (1.f - h1 * h1);
            float w10 = sm[OFF_W1 + n], w11 = sm[OFF_W1 + HID + n], w12 = sm[OFF_W1 + 2*HID + n];
            float Vs = w10 * sm[OFF_V + base + n]
                     + w11 * sm[OFF_V + 1 * SZ_T + base + n]
                     + w12 * sm[OFF_V + 2 * SZ_T + base + n];
            float gm = g1dd * Vs;
            gd0 += gm * w10; gd1 += gm * w11; gd2 += gm * w12;
        }
        float* rd = &sm[OFF_RED + tid * 8];
        rd[0]=lap0; rd[1]=lap1; rd[2]=lap2; rd[3]=gd0; rd[4]=gd1; rd[5]=gd2; rd[6]=zp; rd[7]=zs;
    }
    __syncthreads();

    if ((tid & 7) == 0) {
        const int p = tid >> 3;
        float acc[8] = {0,0,0,0,0,0,0,0};
        for (int s = 0; s < 8; ++s) {                 // fixed order -> deterministic
            const float* rd = &sm[OFF_RED + (p * 8 + s) * 8];
            for (int j = 0; j < 8; ++j) acc[j] += rd[j];
        }
        float zp = acc[6] + bp[0];
        float zs = acc[7] + bs[0];
        float stiff = fmaxf(zs, 0.f) + log1pf(expf(-fabsf(zs)));   // softplus
        const float NU = 0.49f;
        float mu = stiff * (1.f / (2.f * (1.f + NU)));
        float denom = fmaxf(1.f - 2.f * NU, 1e-5f);
        float lm = stiff * (NU / ((1.f + NU) * denom));
        float c2 = lm + mu;
        float phys = 0.f;
        for (int i = 0; i < 3; ++i) {
            float res = mu * acc[i] + c2 * acc[3 + i];
            phys += res * res;
        }
        float pred = 1.f / (1.f + expf(-zp));
        pred = fminf(fmaxf(pred, 1e-7f), 1.f - 1e-7f);
        float tl = fminf(fmaxf(labels[p0 + p], 0.f), 1.f);
        float bce = -(tl * logf(pred) + (1.f - tl) * logf(1.f - pred));
        float d = stiff - tstiff[p0 + p];
        float el = fmaxf(-stiff, 0.f) + fmaxf(stiff - 15.f, 0.f);
        float* ls = &sm[OFF_LOSS + p * 4];
        ls[0] = bce; ls[1] = d * d; ls[2] = phys; ls[3] = el;
    }
    __syncthreads();

    if (tid == 0) {
        float s0=0, s1=0, s2=0, s3=0;
        for (int p = 0; p < PTS; ++p) {
            s0 += sm[OFF_LOSS + p*4 + 0];
            s1 += sm[OFF_LOSS + p*4 + 1];
            s2 += sm[OFF_LOSS + p*4 + 2];
            s3 += sm[OFF_LOSS + p*4 + 3];
        }
        partial[blockIdx.x * 4 + 0] = s0;
        partial[blockIdx.x * 4 + 1] = s1;
        partial[blockIdx.x * 4 + 2] = s2;
        partial[blockIdx.x * 4 + 3] = s3;
    }
}

__global__ void pinn_finalize(const float* __restrict__ partial, int nblocks,
                              float* __restrict__ out, float invB)
{
    __shared__ float ssum[4];
    int c = threadIdx.x;
    if (c < 4) {
        float s = 0.f;
        for (int g = 0; g < nblocks; ++g) s += partial[g * 4 + c];  // fixed order
        ssum[c] = s;
    }
    __syncthreads();
    if (c == 0) {
        float data = ssum[0] * invB + 0.5f * ssum[1] * invB;
        float phys = ssum[2] * invB;
        float el   = ssum[3] * invB;
        float total = data + 0.1f * phys + 0.05f * el;
        out[0] = total; out[1] = data; out[2] = phys; out[3] = el;
    }
}

extern "C" void kernel_launch(void* const* d_in, const int* in_sizes, int n_in,
                              void* d_out, int out_size, void* d_ws, size_t ws_size,
                              hipStream_t stream)
{
    const float* coords = (const float*)d_in[0];
    const float* labels = (const float*)d_in[1];
    const float* tstiff = (const float*)d_in[2];
    const float* W1 = (const float*)d_in[3];
    const float* b1 = (const float*)d_in[4];
    const float* W2 = (const float*)d_in[5];
    const float* b2 = (const float*)d_in[6];
    const float* Wp = (const float*)d_in[7];
    const float* bp = (const float*)d_in[8];
    const float* Ws = (const float*)d_in[9];
    const float* bs = (const float*)d_in[10];
    const float* Wd = (const float*)d_in[11];
    // d_in[12] = bd: unused (only derivatives of u are needed)

    const int Bn = in_sizes[0] / 3;
    const int nblocks = Bn / PTS;          // 32768/16 = 2048
    float* partial = (float*)d_ws;         // nblocks*4 floats = 32 KB

    const size_t smem = SMEM_FLOATS * sizeof(float);
    pinn_main<<<nblocks, NTH, smem, stream>>>(coords, labels, tstiff, W1, b1, W2, b2,
                                              Wp, bp, Ws, bs, Wd, partial, Bn);
    pinn_finalize<<<1, 64, 0, stream>>>(partial, nblocks, (float*)d_out,
                                        1.0f / (float)Bn);
}